// IsingGNN_58591943852343
// MI455X (gfx1250) — compile-verified
//
#include <hip/hip_runtime.h>

typedef __attribute__((ext_vector_type(2))) float v2f;
typedef __attribute__((ext_vector_type(8))) float v8f;

#define LSIDE 512
#define SITES (LSIDE * LSIDE)            // 262144 sites per batch
#define NBATCH 64
#define THREADS 256                      // 8 wave32 per block
#define SITES_PER_THREAD 4               // one lattice-row-aligned group of 4
#define SITES_PER_BLOCK 8192
#define CHUNKS_PER_BATCH (SITES / SITES_PER_BLOCK)            // 32
#define NBLOCKS (NBATCH * CHUNKS_PER_BATCH)                   // 2048 (exact tiling)
#define ITERS (SITES_PER_BLOCK / (THREADS * SITES_PER_THREAD)) // 8 (uniform)

// spin from argmax over K=2: index 1 wins only on strict >, ties -> index 0 -> spin -1
__device__ __forceinline__ float spin_of(float x0, float x1) {
    return (x1 > x0) ? 1.0f : -1.0f;
}

__global__ __launch_bounds__(THREADS)
void ising_energy_partials(const float* __restrict__ x, float* __restrict__ partials)
{
    const int blk   = blockIdx.x;
    const int b     = blk / CHUNKS_PER_BATCH;
    const int chunk = blk % CHUNKS_PER_BATCH;

    const float4* __restrict__ x4 = reinterpret_cast<const float4*>(x);
    const float2* __restrict__ x2 = reinterpret_cast<const float2*>(x);
    const size_t base2 = (size_t)b * SITES;   // batch offset in float2 (per-site) units
    const size_t base4 = base2 >> 1;          // batch offset in float4 units

    // partials (two sub-accumulators each for FMA-chain ILP):
    // p0=Sum(x0+x1), p1=Sum(s*h), p2=Sum((x0+x1)*s*h), p3=Sum((x0-x1)*h)
    float p0a = 0.f, p0b = 0.f, p1a = 0.f, p1b = 0.f;
    float p2a = 0.f, p2b = 0.f, p3a = 0.f, p3b = 0.f;

    int s0 = chunk * SITES_PER_BLOCK + SITES_PER_THREAD * (int)threadIdx.x;
#pragma unroll 2
    for (int it = 0; it < ITERS; ++it, s0 += SITES_PER_THREAD * THREADS) {
        // speculative prefetch of the streaming center line (global_prefetch_b8)
        __builtin_prefetch((const void*)(x4 + base4 + (s0 >> 1) + 1024), 0, 3);

        const int r   = s0 >> 9;           // lattice row (same for all 4 sites)
        const int col = s0 & (LSIDE - 1);  // multiple of 4

        // center: sites s0..s0+3 (4 x float2 = 2 x b128, fully coalesced)
        const float4 c01 = x4[base4 + (s0 >> 1)];
        const float4 c23 = x4[base4 + (s0 >> 1) + 1];

        // vertical neighbors: whole row shifts, shared by the 4 sites
        const int du = (r == 0)         ? (SITES - LSIDE) : -LSIDE;
        const int dd = (r == LSIDE - 1) ? (LSIDE - SITES) :  LSIDE;
        const float4 u01 = x4[base4 + ((s0 + du) >> 1)];
        const float4 u23 = x4[base4 + ((s0 + du) >> 1) + 1];
        const float4 d01 = x4[base4 + ((s0 + dd) >> 1)];
        const float4 d23 = x4[base4 + ((s0 + dd) >> 1) + 1];

        // horizontal halo (interior horizontal spins reused from center regs)
        const int sl = (col == 0)         ? s0 + LSIDE - 1 : s0 - 1;
        const int sr = (col == LSIDE - 4) ? s0 + 4 - LSIDE : s0 + 4;
        const float2 xl = x2[base2 + sl];
        const float2 xr = x2[base2 + sr];

        // spins (14 compares per 4 sites)
        const float sc0 = spin_of(c01.x, c01.y), sc1 = spin_of(c01.z, c01.w);
        const float sc2 = spin_of(c23.x, c23.y), sc3 = spin_of(c23.z, c23.w);
        const float su0 = spin_of(u01.x, u01.y), su1 = spin_of(u01.z, u01.w);
        const float su2 = spin_of(u23.x, u23.y), su3 = spin_of(u23.z, u23.w);
        const float sd0 = spin_of(d01.x, d01.y), sd1 = spin_of(d01.z, d01.w);
        const float sd2 = spin_of(d23.x, d23.y), sd3 = spin_of(d23.z, d23.w);
        const float slp = spin_of(xl.x, xl.y),   srp = spin_of(xr.x, xr.y);

        // local fields
        const float h0 = (slp + sc1) + (su0 + sd0);
        const float h1 = (sc0 + sc2) + (su1 + sd1);
        const float h2 = (sc1 + sc3) + (su2 + sd2);
        const float h3 = (sc2 + srp) + (su3 + sd3);

        const float sum0 = c01.x + c01.y, dif0 = c01.x - c01.y;
        const float sum1 = c01.z + c01.w, dif1 = c01.z - c01.w;
        const float sum2 = c23.x + c23.y, dif2 = c23.x - c23.y;
        const float sum3 = c23.z + c23.w, dif3 = c23.z - c23.w;
        const float sh0 = sc0 * h0, sh1 = sc1 * h1, sh2 = sc2 * h2, sh3 = sc3 * h3;

        p0a += sum0 + sum1;           p0b += sum2 + sum3;
        p1a += sh0 + sh1;             p1b += sh2 + sh3;
        p2a = fmaf(sum0, sh0, fmaf(sum1, sh1, p2a));
        p2b = fmaf(sum2, sh2, fmaf(sum3, sh3, p2b));
        p3a = fmaf(dif0, h0,  fmaf(dif1, h1,  p3a));
        p3b = fmaf(dif2, h2,  fmaf(dif3, h3,  p3b));
    }

    const float p0 = p0a + p0b, p1 = p1a + p1b, p2 = p2a + p2b, p3 = p3a + p3b;

    // ---- wave32 reduction via V_WMMA_F32_16X16X4_F32 (full f32 precision) ----
    // A layout (16x4 f32): lane m (0-15): vgpr0=A[m,0], vgpr1=A[m,1];
    //                      lane 16+m:    vgpr0=A[m,2], vgpr1=A[m,3].
    // B layout (4x16 f32): vgpr0: lanes0-15 = row K=0, lanes16-31 = K=2;
    //                      vgpr1: lanes0-15 = row K=1, lanes16-31 = K=3.
    // One-hot B columns route quantity q's lane-pair sums into C column q.
    const unsigned lane = threadIdx.x & 31u;
    const unsigned wave = threadIdx.x >> 5;

    v2f a0; a0.x = p0; a0.y = p1;
    v2f b0; b0.x = (lane == 0u || lane == 16u) ? 1.0f : 0.0f;   // B[0,0], B[2,0]
            b0.y = (lane == 1u || lane == 17u) ? 1.0f : 0.0f;   // B[1,1], B[3,1]
    v8f c = {};
    c = __builtin_amdgcn_wmma_f32_16x16x4_f32(false, a0, false, b0,
                                              (short)0, c, false, false);
    v2f a1; a1.x = p2; a1.y = p3;
    v2f b1; b1.x = (lane == 2u || lane == 18u) ? 1.0f : 0.0f;   // B[0,2], B[2,2]
            b1.y = (lane == 3u || lane == 19u) ? 1.0f : 0.0f;   // B[1,3], B[3,3]
    c = __builtin_amdgcn_wmma_f32_16x16x4_f32(false, a1, false, b1,
                                              (short)0, c, false, false);

    // C layout: lane n holds C[0..7, n]; lane 16+n holds C[8..15, n].
    // Column q lives entirely in lanes q and 16+q -> 8-float serial add per lane.
    const float rsum = ((c[0] + c[1]) + (c[2] + c[3])) + ((c[4] + c[5]) + (c[6] + c[7]));

    __shared__ float red[8][8];
    if (lane < 4u)                 red[wave][lane] = rsum;               // lower half of quantity `lane`
    if (lane >= 16u && lane < 20u) red[wave][4u + (lane - 16u)] = rsum;  // upper half
    __syncthreads();

    if (threadIdx.x < 4u) {
        float tot = 0.0f;
        for (int w = 0; w < 8; ++w)
            tot += red[w][threadIdx.x] + red[w][4 + threadIdx.x];
        partials[(size_t)blk * 4 + threadIdx.x] = tot;
    }
}

__global__ void ising_finalize(const float* __restrict__ partials,
                               float* __restrict__ out)
{
    const int b = threadIdx.x;
    if (b >= NBATCH) return;
    // chunks of batch b are contiguous: blocks [b*32, b*32+32)
    const float* p = partials + (size_t)b * CHUNKS_PER_BATCH * 4;
    float P0 = 0.0f, P1 = 0.0f, P2 = 0.0f, P3 = 0.0f;
    for (int ch = 0; ch < CHUNKS_PER_BATCH; ++ch) {   // fixed order -> deterministic
        P0 += p[ch * 4 + 0];
        P1 += p[ch * 4 + 1];
        P2 += p[ch * 4 + 2];
        P3 += p[ch * 4 + 3];
    }
    const float total = -0.5f * P1 * P0 + P2 + P3;
    out[b] = total / (float)SITES;
}

extern "C" void kernel_launch(void* const* d_in, const int* in_sizes, int n_in,
                              void* d_out, int out_size, void* d_ws, size_t ws_size,
                              hipStream_t stream) {
    (void)in_sizes; (void)n_in; (void)out_size; (void)ws_size;
    const float* x  = (const float*)d_in[0];   // [64, 262144, 2] f32
    float* out      = (float*)d_out;           // [64] f32
    float* partials = (float*)d_ws;            // 2048 blocks * 4 floats = 32 KB

    ising_energy_partials<<<NBLOCKS, THREADS, 0, stream>>>(x, partials);
    ising_finalize<<<1, 64, 0, stream>>>(partials, out);
}